// SiluAndMul_84920093376533
// MI455X (gfx1250) — compile-verified
//
#include <hip/hip_runtime.h>

// SiluAndMul: out[m, n] = silu(x[m, n]) * x[m, n + N/2]
//   x:   fp32 [M, 22016]
//   out: fp32 [M, 11008]
//
// Memory-bound streaming kernel (~2.16 GB round trip -> ~93 us at 23.3 TB/s).
// Strategy: float4 (b128) non-temporal loads/stores, 2 float4 per thread for
// MLP depth, wave32-friendly 256-thread blocks, no LDS (zero reuse op).

typedef float v4f __attribute__((ext_vector_type(4)));

#define N_FULL    22016
#define N_HALF    11008
#define VEC_ROW   (N_HALF / 4)      // 2752 float4 per output row
#define TPB       256
#define VPT       2                 // float4 elements per thread
#define TILE      (TPB * VPT)       // 512 float4 per block

__device__ __forceinline__ float silu_f(float g) {
    // g * sigmoid(g) = g / (1 + exp(-g)); v_exp_f32 + v_rcp_f32
    return g * __builtin_amdgcn_rcpf(1.0f + __expf(-g));
}

__global__ __launch_bounds__(TPB) void SiluAndMul_84920093376533_kernel(
    const float* __restrict__ x, float* __restrict__ out) {
    const int row = blockIdx.y;
    const int c0  = blockIdx.x * TILE + threadIdx.x;

    const v4f* __restrict__ gate =
        (const v4f*)(x + (size_t)row * N_FULL);
    const v4f* __restrict__ up =
        (const v4f*)(x + (size_t)row * N_FULL + N_HALF);
    v4f* __restrict__ o = (v4f*)(out + (size_t)row * N_HALF);

#pragma unroll
    for (int k = 0; k < VPT; ++k) {
        const int c = c0 + k * TPB;
        if (c < VEC_ROW) {
            // Streamed once, >> L2 (192 MB): non-temporal both ways.
            v4f g = __builtin_nontemporal_load(&gate[c]);
            v4f u = __builtin_nontemporal_load(&up[c]);
            v4f r;
            r.x = silu_f(g.x) * u.x;
            r.y = silu_f(g.y) * u.y;
            r.z = silu_f(g.z) * u.z;
            r.w = silu_f(g.w) * u.w;
            __builtin_nontemporal_store(r, &o[c]);
        }
    }
}

extern "C" void kernel_launch(void* const* d_in, const int* in_sizes, int n_in,
                              void* d_out, int out_size, void* d_ws, size_t ws_size,
                              hipStream_t stream) {
    const float* x = (const float*)d_in[0];
    float* out     = (float*)d_out;

    const int rows = in_sizes[0] / N_FULL;            // 16384 for the reference shape
    dim3 grid((VEC_ROW + TILE - 1) / TILE, rows);     // (6, 16384)
    dim3 block(TPB);

    SiluAndMul_84920093376533_kernel<<<grid, block, 0, stream>>>(x, out);
}